// ABC_2D_76656576299098
// MI455X (gfx1250) — compile-verified
//
#include <hip/hip_runtime.h>

typedef __attribute__((ext_vector_type(2))) float v2f;
typedef __attribute__((ext_vector_type(8))) float v8f;

#define BATCH    16
#define CCH      64
#define HH       128
#define WW       128
#define HWP      (HH * WW)          // 16384 pixels
#define CHW      (CCH * HWP)        // 1,048,576 elems per batch image
#define KC       576                // K*C (9*64)
#define KN       64                 // output channels
#define ST       580                // padded LDS row stride in dwords (bank-conflict-free)
#define PIX_TILE 16
#define THREADS  512                // 16 waves (wave32) -> one wave per batch

__global__ __launch_bounds__(THREADS)
void hashconv_wmma_f32(const float* __restrict__ x,
                       const long long* __restrict__ ht,
                       const float* __restrict__ wgt,
                       float* __restrict__ out) {
    // 64*580*4 = 148,480 B weights + 16*580*4 = 37,120 B offsets = 185,600 B
    // (CDNA5 WGP has 320 KB LDS; one workgroup resident -> 4 waves/SIMD32)
    __shared__ float w_lds[KN * ST];
    __shared__ int   i_lds[PIX_TILE * ST];

    const int tid   = threadIdx.x;
    const int pbase = blockIdx.x * PIX_TILE;

    // ---- stage weights into LDS (row-major, padded stride) ----
    for (int i = tid; i < KN * KC; i += THREADS) {
        int r = i / KC;
        int c = i - r * KC;
        w_lds[r * ST + c] = wgt[i];
    }
    // ---- stage per-pixel gather indices (kept as element indices: lets the
    //      compiler use SADDR + 32-bit VGPR offset w/ scale for the gathers) ----
    for (int i = tid; i < PIX_TILE * KC; i += THREADS) {
        int n = i / KC;
        int q = i - n * KC;
        i_lds[n * ST + q] = (int)ht[(size_t)(pbase + n) * KC + q];
    }
    __syncthreads();

    const int lane  = tid & 31;
    const int b     = tid >> 5;        // wave id == batch id
    const int n     = lane & 15;       // A: row M; B: col N (pixel); same lane role
    const int khalf = lane >> 4;       // 0: K pair {0,1}, 1: K pair {2,3}

    const float* __restrict__ xb = x + (size_t)b * CHW;   // uniform per wave

    v8f acc0 = {}, acc1 = {}, acc2 = {}, acc3 = {};

    const int*   iptr = &i_lds[n * ST];
    const float* w0   = &w_lds[( 0 + n) * ST];
    const float* w1   = &w_lds[(16 + n) * ST];
    const float* w2   = &w_lds[(32 + n) * ST];
    const float* w3   = &w_lds[(48 + n) * ST];

    for (int kbase = 0; kbase < KC; kbase += 4) {
        const int koff = kbase + 2 * khalf;

        // two gather indices for this lane's pixel (one ds_load_b64)
        int2 idx2 = *(const int2*)(iptr + koff);
        // the gathers: divergent 4B loads from the L2-resident batch image
        float g0 = xb[(unsigned)idx2.x];
        float g1 = xb[(unsigned)idx2.y];
        v2f bf = { g0, g1 };   // B 4x16: V0 -> K=2*khalf, V1 -> K=2*khalf+1, N=n

        // A 16x4 fragments from LDS: V0 -> K=2*khalf, V1 -> K=2*khalf+1, M=n
        v2f a0 = *(const v2f*)(w0 + koff);
        v2f a1 = *(const v2f*)(w1 + koff);
        v2f a2 = *(const v2f*)(w2 + koff);
        v2f a3 = *(const v2f*)(w3 + koff);

        acc0 = __builtin_amdgcn_wmma_f32_16x16x4_f32(false, a0, false, bf, (short)0, acc0, false, false);
        acc1 = __builtin_amdgcn_wmma_f32_16x16x4_f32(false, a1, false, bf, (short)0, acc1, false, false);
        acc2 = __builtin_amdgcn_wmma_f32_16x16x4_f32(false, a2, false, bf, (short)0, acc2, false, false);
        acc3 = __builtin_amdgcn_wmma_f32_16x16x4_f32(false, a3, false, bf, (short)0, acc3, false, false);
    }

    // ---- epilogue: C/D 16x16: VGPR v holds M=v (lanes 0-15), M=v+8 (lanes 16-31).
    // Fold all lane-variant terms into one base pointer so the 32 stores use
    // immediate offsets (max 55*HWP*4 = 3.6MB < 24-bit signed ioffset range).
    const int kn_off = khalf ? 8 : 0;
    float* __restrict__ ob =
        out + (size_t)b * KN * HWP + (size_t)kn_off * HWP + (unsigned)(pbase + n);
#pragma unroll
    for (int v = 0; v < 8; ++v) {
        ob[(size_t)( 0 + v) * HWP] = acc0[v];
        ob[(size_t)(16 + v) * HWP] = acc1[v];
        ob[(size_t)(32 + v) * HWP] = acc2[v];
        ob[(size_t)(48 + v) * HWP] = acc3[v];
    }
}

extern "C" void kernel_launch(void* const* d_in, const int* in_sizes, int n_in,
                              void* d_out, int out_size, void* d_ws, size_t ws_size,
                              hipStream_t stream) {
    const float*     x  = (const float*)d_in[0];
    const long long* ht = (const long long*)d_in[1];
    const float*     w  = (const float*)d_in[2];
    float*           o  = (float*)d_out;

    hipLaunchKernelGGL(hashconv_wmma_f32,
                       dim3(HWP / PIX_TILE), dim3(THREADS), 0, stream,
                       x, ht, w, o);
}